// SparseResUNet_56392920596829
// MI455X (gfx1250) — compile-verified
//
#include <hip/hip_runtime.h>
#include <stdint.h>

typedef __attribute__((ext_vector_type(2))) float v2f;
typedef __attribute__((ext_vector_type(8))) float v8f;

// ---------------------------------------------------------------------------
// Kernels
// ---------------------------------------------------------------------------

__global__ void k_zero(float* __restrict__ p, long long n) {
  long long i = (long long)blockIdx.x * blockDim.x + threadIdx.x;
  long long st = (long long)gridDim.x * blockDim.x;
  for (; i < n; i += st) p[i] = 0.f;
}

// Sparse conv for one (offset k, 16-row tile): async-gather A into LDS
// (global_load_async_to_lds_b128, ASYNCcnt), WMMA f32 16x16x4 over Cin,
// scatter-add via rulebook with global f32 atomics.
// rin/rout == nullptr => dense GEMM (identity rulebook), used for 1x1 shortcut.
__global__ void k_spconv_wmma(const float* __restrict__ x,
                              const float* __restrict__ W,
                              const int* __restrict__ rin,
                              const int* __restrict__ rout,
                              int M, int Cin, int Cout,
                              float* __restrict__ acc, int mtiles) {
  extern __shared__ char smem[];
  const int lda = Cin + 4;  // rows stay 16B-aligned; stride%64 banks conflict-free
  float* sA = (float*)smem;
  int* rIn = (int*)(smem + (size_t)16 * lda * sizeof(float));
  int* rOut = rIn + 16;

  const int k = blockIdx.x / mtiles;
  const int mt = blockIdx.x - k * mtiles;
  const int tid = threadIdx.x;

  if (tid < 16) {
    int m = mt * 16 + tid;
    int ri = -1, ro = -1;
    if (m < M) {
      ri = rin ? rin[(long long)k * M + m] : m;
      ro = rout ? rout[(long long)k * M + m] : m;
    }
    rIn[tid] = ri;
    rOut[tid] = ro;
  }
  __syncthreads();

  // Rulebook gather: 16B per lane, global -> LDS without touching VGPRs.
  const int cpr = Cin >> 2;  // float4 chunks per row
  for (int idx = tid; idx < 16 * cpr; idx += blockDim.x) {
    int m = idx / cpr;
    int c4 = idx - m * cpr;
    int r = rIn[m];
    float* dstp = sA + m * lda + (c4 << 2);
    if (r >= 0) {
      unsigned ldsoff = (unsigned)(uintptr_t)dstp;
      unsigned long long ga =
          (unsigned long long)(uintptr_t)(x + (long long)r * Cin + (c4 << 2));
      asm volatile("global_load_async_to_lds_b128 %0, %1, off"
                   :: "v"(ldsoff), "v"(ga)
                   : "memory");
    } else {
      float4 z = {0.f, 0.f, 0.f, 0.f};
      *(float4*)dstp = z;  // pad rows (tail tile) -> plain LDS store
    }
  }
  asm volatile("s_wait_asynccnt 0x0" ::: "memory");
  __syncthreads();

  const int wave = tid >> 5;
  const int lane = tid & 31;
  const int half = lane >> 4;   // 0: K pair {0,1}, 1: K pair {2,3}
  const int lr = lane & 15;     // M index for A frag, N index for B/C frags
  const float* __restrict__ Wk = W + (long long)k * Cin * Cout;
  const int ntiles = Cout >> 4;
  const int nwaves = blockDim.x >> 5;

  for (int nt = wave; nt < ntiles; nt += nwaves) {
    const int ncol = (nt << 4) + lr;
    v8f c = {0.f, 0.f, 0.f, 0.f, 0.f, 0.f, 0.f, 0.f};
    for (int kk = 0; kk < Cin; kk += 4) {
      const int k0 = kk + (half ? 2 : 0);
      v2f a, b;
      // A 16x4 (MxK): lanes 0-15 hold K={kk,kk+1}, lanes 16-31 hold K={kk+2,kk+3}
      a.x = sA[lr * lda + k0];
      a.y = sA[lr * lda + k0 + 1];
      // B 4x16 (KxN): same K striping, N = lane 0..15 within each half
      b.x = Wk[(long long)k0 * Cout + ncol];
      b.y = Wk[(long long)(k0 + 1) * Cout + ncol];
      c = __builtin_amdgcn_wmma_f32_16x16x4_f32(false, a, false, b,
                                                (short)0, c, false, false);
    }
    // D 16x16 f32: lane lr holds rows {rbase..rbase+7} at column ncol
    const int rbase = half ? 8 : 0;
#pragma unroll
    for (int r = 0; r < 8; ++r) {
      int ro = rOut[rbase + r];
      if (ro >= 0) atomicAdd(&acc[(long long)ro * Cout + ncol], c[r]);
    }
  }
}

// Per-channel sum / sumsq reduction. blockDim.x == C (multiple of 32).
__global__ void k_stats(const float* __restrict__ acc, int n, int rows_per_blk,
                        float* __restrict__ st) {
  const int C = blockDim.x;
  const int c = threadIdx.x;
  int r0 = blockIdx.x * rows_per_blk;
  int r1 = r0 + rows_per_blk;
  if (r1 > n) r1 = n;
  float s = 0.f, s2 = 0.f;
  for (int r = r0; r < r1; ++r) {
    float v = acc[(long long)r * C + c];
    s += v;
    s2 += v * v;
  }
  atomicAdd(&st[c], s);
  atomicAdd(&st[C + c], s2);
}

// y = BN(acc) [+ resid] [ReLU], written to dst with row stride ldd at coff.
__global__ void k_bn_apply(const float* __restrict__ acc, long long n, int C,
                           const float* __restrict__ st,
                           const float* __restrict__ g,
                           const float* __restrict__ be,
                           const float* __restrict__ resid,
                           float* __restrict__ dst, int ldd, int coff,
                           int relu, float inv_n) {
  const long long total = n * C;
  long long i = (long long)blockIdx.x * blockDim.x + threadIdx.x;
  const long long stp = (long long)gridDim.x * blockDim.x;
  for (; i < total; i += stp) {
    const int c = (int)(i % C);
    const long long r = i / C;
    const float mean = st[c] * inv_n;
    const float var = st[C + c] * inv_n - mean * mean;
    float y = (acc[i] - mean) * rsqrtf(var + 1e-5f) * g[c] + be[c];
    if (resid) y += resid[i];
    if (relu) y = fmaxf(y, 0.f);
    dst[r * ldd + coff + c] = y;
  }
}

// Copy packed [n,Cs] into strided dst columns [coff, coff+Cs) (skip concat).
__global__ void k_copy_cols(const float* __restrict__ src, int Cs,
                            float* __restrict__ dst, int ldd, int coff,
                            long long n) {
  const long long total = n * Cs;
  long long i = (long long)blockIdx.x * blockDim.x + threadIdx.x;
  const long long stp = (long long)gridDim.x * blockDim.x;
  for (; i < total; i += stp) {
    const int c = (int)(i % Cs);
    const long long r = i / Cs;
    dst[r * ldd + coff + c] = src[i];
  }
}

// ---------------------------------------------------------------------------
// Host orchestration
// ---------------------------------------------------------------------------

extern "C" void kernel_launch(void* const* d_in, const int* in_sizes, int n_in,
                              void* d_out, int out_size, void* d_ws,
                              size_t ws_size, hipStream_t stream) {
  (void)in_sizes; (void)n_in; (void)out_size; (void)ws_size;
  static const int N_LEV[5] = {120000, 48000, 19200, 7680, 3072};
  static const int CH[9] = {32, 32, 64, 128, 256, 256, 128, 96, 96};

  auto F = [&](int i) { return (const float*)d_in[i]; };
  auto I = [&](int i) { return (const int*)d_in[i]; };

  struct Conv { const float *W, *g, *be; };
  struct Res { Conv c1, c2, sc; bool has_sc; };

  int ip = 1;  // d_in[0] == feats
  auto conv = [&]() { Conv c{F(ip), F(ip + 2), F(ip + 3)}; ip += 4; return c; };
  auto res = [&](int ci, int co) {
    Res r; r.c1 = conv(); r.c2 = conv();
    r.has_sc = (ci != co);
    if (r.has_sc) r.sc = conv();
    return r;
  };

  Conv stem0 = conv(), stem1 = conv();
  Conv encd[4]; Res encr0[4], encr1[4];
  for (int k = 0; k < 4; ++k) {
    encd[k] = conv();
    encr0[k] = res(CH[k], CH[k + 1]);
    encr1[k] = res(CH[k + 1], CH[k + 1]);
  }
  Conv decu[4]; Res decr0[4], decr1[4];
  for (int k = 0; k < 4; ++k) {
    decu[k] = conv();
    decr0[k] = res(CH[k + 5] + CH[3 - k], CH[k + 5]);
    decr1[k] = res(CH[k + 5], CH[k + 5]);
  }
  const int *si[5], *so[5], *di[4], *dd[4];
  for (int i = 0; i < 5; ++i) si[i] = I(ip++);
  for (int i = 0; i < 5; ++i) so[i] = I(ip++);
  for (int i = 0; i < 4; ++i) di[i] = I(ip++);
  for (int i = 0; i < 4; ++i) dd[i] = I(ip++);

  // Workspace arena (floats)
  float* ws = (float*)d_ws;
  float* stA = ws;                        // 512
  float* stB = ws + 512;                  // 512
  float* bufA = ws + 1024;                // conv accum      [<=120000*96]
  float* bufB = bufA + 11520000LL;        // shortcut accum  [<=120000*96]
  float* bufM = bufB + 11520000LL;        // m / s           [<=120000*96]
  float* bufZ = bufM + 11520000LL;        // concat / r      [<=120000*128]
  float* bufY = bufZ + 15360000LL;        // d / y1          [<=120000*96]

  auto zero = [&](float* p, long long n) {
    long long g = (n + 255) / 256;
    if (g > 65536) g = 65536;
    k_zero<<<dim3((unsigned)g), dim3(256), 0, stream>>>(p, n);
  };
  auto spconv = [&](const float* x, const Conv& cv, const int* rin,
                    const int* rout, int K, int M, int Cin, int Cout,
                    float* acc, int n_out) {
    zero(acc, (long long)n_out * Cout);
    const int mtiles = (M + 15) / 16;
    const size_t sh = (size_t)16 * (Cin + 4) * sizeof(float) + 32 * sizeof(int);
    k_spconv_wmma<<<dim3((unsigned)(K * mtiles)), dim3(128), sh, stream>>>(
        x, cv.W, rin, rout, M, Cin, Cout, acc, mtiles);
  };
  auto stats = [&](const float* acc, int n, int C, float* st) {
    zero(st, 2LL * C);
    const int rpb = 256;
    const int g = (n + rpb - 1) / rpb;
    k_stats<<<dim3((unsigned)g), dim3((unsigned)C), 0, stream>>>(acc, n, rpb, st);
  };
  auto bnapply = [&](const float* acc, int n, int C, const float* st,
                     const Conv& cv, const float* resid, float* dst, int ldd,
                     int coff, int relu) {
    long long g = ((long long)n * C + 255) / 256;
    if (g > 65536) g = 65536;
    k_bn_apply<<<dim3((unsigned)g), dim3(256), 0, stream>>>(
        acc, (long long)n, C, st, cv.g, cv.be, resid, dst, ldd, coff, relu,
        1.0f / (float)n);
  };
  auto cbr = [&](const float* x, const Conv& cv, const int* rin,
                 const int* rout, int K, int M, int Cin, int Cout, int n_out,
                 float* dst, int ldd, int coff) {
    spconv(x, cv, rin, rout, K, M, Cin, Cout, bufA, n_out);
    stats(bufA, n_out, Cout, stA);
    bnapply(bufA, n_out, Cout, stA, cv, nullptr, dst, ldd, coff, 1);
  };
  auto resblock = [&](const float* x, const Res& r, const int* rin,
                      const int* rout, int M, int n, int Cin, int Cout,
                      float* dst) {
    spconv(x, r.c1, rin, rout, 27, M, Cin, Cout, bufA, n);
    stats(bufA, n, Cout, stA);
    bnapply(bufA, n, Cout, stA, r.c1, nullptr, bufM, Cout, 0, 1);
    spconv(bufM, r.c2, rin, rout, 27, M, Cout, Cout, bufA, n);
    stats(bufA, n, Cout, stA);
    const float* resid = x;
    if (r.has_sc) {
      spconv(x, r.sc, nullptr, nullptr, 1, n, Cin, Cout, bufB, n);
      stats(bufB, n, Cout, stB);
      bnapply(bufB, n, Cout, stB, r.sc, nullptr, bufM, Cout, 0, 0);
      resid = bufM;  // m already consumed by c2
    }
    bnapply(bufA, n, Cout, stA, r.c2, resid, dst, Cout, 0, 1);
  };

  // Output tuple layout: x0..x4, then decoder outputs (level 3,2,1,0)
  float* out = (float*)d_out;
  float* x0 = out;
  float* x1 = x0 + 120000LL * 32;
  float* x2 = x1 + 48000LL * 32;
  float* x3 = x2 + 19200LL * 64;
  float* x4 = x3 + 7680LL * 128;
  float* y3 = x4 + 3072LL * 256;
  float* y2 = y3 + 7680LL * 256;
  float* y1 = y2 + 19200LL * 128;
  float* y0 = y1 + 48000LL * 96;

  int Ms[5], Md[4];
  for (int i = 0; i < 5; ++i) Ms[i] = N_LEV[i] / 3;
  for (int i = 0; i < 4; ++i) Md[i] = N_LEV[i] / 8;

  // Stem
  cbr(F(0), stem0, si[0], so[0], 27, Ms[0], 4, 32, N_LEV[0], bufY, 32, 0);
  cbr(bufY, stem1, si[0], so[0], 27, Ms[0], 32, 32, N_LEV[0], x0, 32, 0);

  // Encoder
  float* xs[5] = {x0, x1, x2, x3, x4};
  for (int k = 0; k < 4; ++k) {
    const int n = N_LEV[k + 1];
    cbr(xs[k], encd[k], di[k], dd[k], 8, Md[k], CH[k], CH[k], n, bufY, CH[k], 0);
    resblock(bufY, encr0[k], si[k + 1], so[k + 1], Ms[k + 1], n, CH[k],
             CH[k + 1], bufZ);
    resblock(bufZ, encr1[k], si[k + 1], so[k + 1], Ms[k + 1], n, CH[k + 1],
             CH[k + 1], xs[k + 1]);
  }

  // Decoder
  float* ydst[4] = {y3, y2, y1, y0};
  const float* y = x4;
  for (int k = 0; k < 4; ++k) {
    const int lf = 3 - k;
    const int n = N_LEV[lf];
    const int Cup = CH[k + 5];
    const int Cskip = CH[lf];
    const int Cz = Cup + Cskip;
    const int Cyin = CH[k + 4];
    // inverse conv: transpose of the downsample rulebook
    cbr(y, decu[k], dd[lf], di[lf], 8, Md[lf], Cyin, Cup, n, bufZ, Cz, 0);
    {  // concat skip into bufZ columns [Cup, Cz)
      long long g = ((long long)n * Cskip + 255) / 256;
      if (g > 65536) g = 65536;
      k_copy_cols<<<dim3((unsigned)g), dim3(256), 0, stream>>>(
          xs[lf], Cskip, bufZ, Cz, Cup, (long long)n);
    }
    resblock(bufZ, decr0[k], si[lf], so[lf], Ms[lf], n, Cz, Cup, bufY);
    resblock(bufY, decr1[k], si[lf], so[lf], Ms[lf], n, Cup, Cup, ydst[k]);
    y = ydst[k];
  }
}